// Type1MoEProjector_6227702579639
// MI455X (gfx1250) — compile-verified
//
#include <hip/hip_runtime.h>
#include <hip/hip_bf16.h>

#define B_ 16
#define L_ 3
#define N_ 512
#define D_ 1024
#define H_ 4096
#define E_ 3
#define S_ (B_*L_*N_)      /* 24576 tokens */
#define TPB_ (L_*N_)       /* 1536 tokens per batch */
#define CAP_ (S_/E_)       /* 8192 capacity per expert */
#define CH_ 256            /* tokens per combine chunk (divides TPB_) */
#define NCH_ (S_/CH_)      /* 96 chunks */
#define CPB_ (TPB_/CH_)    /* 6 chunks per batch */

typedef __attribute__((ext_vector_type(2))) float v2f;
typedef __attribute__((ext_vector_type(8))) float v8f;

// token t (in x = transpose(0,2,1,3) order) -> element offset in original tensors
__device__ __forceinline__ long xbase(int t) {
    int b = t / TPB_;
    int r = t - b * TPB_;
    int n = r / L_;
    int l = r - n * L_;
    return ((long)((b * L_ + l) * N_ + n)) * D_;
}

// ---------------- Kernel 1: gating (1 wave per token) ----------------
__global__ void k_gate(const float* __restrict__ x, const float* __restrict__ wg,
                       int* __restrict__ idxA, float* __restrict__ gateA,
                       float* __restrict__ gates3, int* __restrict__ chunkCnt) {
    __shared__ float swg[D_ * E_];
    int tid = threadIdx.x;
    for (int i = tid; i < D_ * E_; i += 256) swg[i] = wg[i];
    __syncthreads();
    int wave = tid >> 5, lane = tid & 31;
    int t = blockIdx.x * 8 + wave;
    const float* xp = x + xbase(t);
    float a0 = 0.f, a1 = 0.f, a2 = 0.f;
    #pragma unroll 8
    for (int i = 0; i < D_ / 32; ++i) {
        int d = lane + i * 32;          // coalesced 128B per wave iter
        float xv = xp[d];
        a0 += xv * swg[d * 3 + 0];
        a1 += xv * swg[d * 3 + 1];
        a2 += xv * swg[d * 3 + 2];
    }
    #pragma unroll
    for (int off = 16; off; off >>= 1) {
        a0 += __shfl_xor(a0, off, 32);
        a1 += __shfl_xor(a1, off, 32);
        a2 += __shfl_xor(a2, off, 32);
    }
    if (lane == 0) {
        float m = fmaxf(a0, fmaxf(a1, a2));
        float g0 = expf(a0 - m), g1 = expf(a1 - m), g2 = expf(a2 - m);
        float inv = 1.0f / (g0 + g1 + g2);
        g0 *= inv; g1 *= inv; g2 *= inv;
        int e = 0; float g = g0;
        if (g1 > g) { g = g1; e = 1; }
        if (g2 > g) { g = g2; e = 2; }
        idxA[t] = e;
        gateA[t] = g;
        gates3[t * 3 + 0] = g0;
        gates3[t * 3 + 1] = g1;
        gates3[t * 3 + 2] = g2;
        atomicAdd(&chunkCnt[(t / CH_) * E_ + e], 1);   // int atomic: deterministic
    }
}

// ---------------- Kernel 2a: deterministic gate-sum reduction ----------------
__global__ void k_gred(const float* __restrict__ gates3, float* __restrict__ gateSum) {
    __shared__ float sred[256];
    int e = blockIdx.x, tid = threadIdx.x;
    float s = 0.f;
    for (int t = tid; t < S_; t += 256) s += gates3[t * 3 + e];
    sred[tid] = s;
    __syncthreads();
    for (int k = 128; k; k >>= 1) { if (tid < k) sred[tid] += sred[tid + k]; __syncthreads(); }
    if (tid == 0) gateSum[e] = sred[0];
}

// ---------------- Kernel 2b: chunk-count scan + l_aux ----------------
__global__ void k_scan(const int* __restrict__ chunkCnt, int* __restrict__ chunkOff,
                       const float* __restrict__ gateSum, float* __restrict__ out_laux) {
    __shared__ int tot[E_];
    int e = threadIdx.x;
    if (e < E_) {
        int run = 0;
        for (int c = 0; c < NCH_; ++c) {
            chunkOff[c * E_ + e] = run;
            run += chunkCnt[c * E_ + e];
        }
        tot[e] = run;
    }
    __syncthreads();
    if (threadIdx.x == 0) {
        float l = 0.f;
        for (int i = 0; i < E_; ++i)
            l += (gateSum[i] / (float)S_) * ((float)tot[i] / (float)S_);
        out_laux[0] = l * (float)E_;
    }
}

// ---------------- Kernel 3: per-batch unmasked count ----------------
__global__ void k_cnt(const int* __restrict__ mask, float* __restrict__ cntB) {
    __shared__ int sred[256];
    int b = blockIdx.x, tid = threadIdx.x;
    int c = 0;
    for (int i = tid; i < TPB_; i += 256)
        c += (mask[b * TPB_ + i] == 0) ? 1 : 0;
    sred[tid] = c;
    __syncthreads();
    for (int s = 128; s; s >>= 1) { if (tid < s) sred[tid] += sred[tid + s]; __syncthreads(); }
    if (tid == 0) cntB[b] = (float)(sred[0] < 1 ? 1 : sred[0]);
}

// ---------------- Kernel 4: combine -> per-chunk z partials ----------------
__global__ void k_combine(const float* __restrict__ x,
                          const int* __restrict__ idxA, const float* __restrict__ gateA,
                          const int* __restrict__ chunkOff,
                          float* __restrict__ zC, float* __restrict__ gsumC) {
    __shared__ float zsh[E_ * D_];
    __shared__ int   sIdx[CH_];
    __shared__ float sGate[CH_];
    int tid = threadIdx.x, blk = blockIdx.x;
    int t0 = blk * CH_;
    for (int i = tid; i < E_ * D_; i += 256) zsh[i] = 0.0f;
    sIdx[tid]  = idxA[t0 + tid];
    sGate[tid] = gateA[t0 + tid];
    __syncthreads();
    // arrival-order rank within chunk; drop tokens past capacity
    {
        int e = sIdx[tid];
        int rank = 0;
        for (int j = 0; j < tid; ++j) rank += (sIdx[j] == e) ? 1 : 0;
        int loc = chunkOff[blk * E_ + e] + rank;
        if (loc >= CAP_) sGate[tid] = 0.0f;   // keep==0
    }
    __syncthreads();
    // race-free accumulation: thread tid owns dims {tid, tid+256, tid+512, tid+768}
    for (int tok = 0; tok < CH_; ++tok) {
        float g = sGate[tok];                 // uniform across block
        if (g == 0.0f) continue;
        int e = sIdx[tok];
        const float* xp = x + xbase(t0 + tok);
        float* zr = zsh + e * D_;
        #pragma unroll
        for (int k = 0; k < 4; ++k) {
            int d = tid + k * 256;            // coalesced global load
            zr[d] += g * xp[d];
        }
    }
    __syncthreads();
    for (int i = tid; i < E_ * D_; i += 256)
        zC[(long)blk * (E_ * D_) + i] = zsh[i];
    if (tid < E_) {
        float s = 0.f;
        for (int j = 0; j < CH_; ++j)
            if (sIdx[j] == tid) s += sGate[j];
        gsumC[blk * E_ + tid] = s;
    }
}

// ---------------- Kernel 5: fixed-order chunk reduction -> z, gsum ----------------
__global__ void k_zred(const float* __restrict__ zC, const float* __restrict__ gsumC,
                       float* __restrict__ z, float* __restrict__ gsum) {
    int be = blockIdx.x;                // b*E_ + e
    int b = be / E_, e = be - b * E_;
    int tid = threadIdx.x;
    for (int d = tid; d < D_; d += 256) {
        float s = 0.f;
        #pragma unroll
        for (int j = 0; j < CPB_; ++j)
            s += zC[((long)(b * CPB_ + j) * E_ + e) * D_ + d];
        z[(b * E_ + e) * D_ + d] = s;
    }
    if (tid == 0) {
        float s = 0.f;
        for (int j = 0; j < CPB_; ++j) s += gsumC[(b * CPB_ + j) * E_ + e];
        gsum[b * E_ + e] = s;
    }
}

// ---------------- Kernel 6: expert GEMM via f32 WMMA: u = sum_e z_e @ We_e + gsum*be ----------------
// M=16 (batch), K=1024 per expert, N=1024. One wave per 16-col tile.
__global__ void k_expert(const float* __restrict__ z, const float* __restrict__ We,
                         const float* __restrict__ be, const float* __restrict__ gsum,
                         float* __restrict__ u) {
    int tid = threadIdx.x;
    int wave = tid >> 5, lane = tid & 31;
    int ml = lane & 15, half = lane >> 4;
    int n0 = (blockIdx.x * 4 + wave) * 16;
    v8f c = {};
    for (int e = 0; e < E_; ++e) {
        const float* zr = z + (ml * E_ + e) * D_;       // A row M=ml of expert e
        const float* wb = We + (long)e * D_ * D_;       // B: [K][N]
        for (int k = 0; k < D_; k += 4) {
            int ka = k + 2 * half;                      // lanes 0-15: K={k,k+1}; 16-31: K={k+2,k+3}
            v2f a, bm;
            a.x  = zr[ka];
            a.y  = zr[ka + 1];
            bm.x = wb[(long)ka * D_ + n0 + ml];
            bm.y = wb[(long)(ka + 1) * D_ + n0 + ml];
            c = __builtin_amdgcn_wmma_f32_16x16x4_f32(false, a, false, bm,
                                                      (short)0, c, false, false);
        }
    }
    int n = n0 + ml;
    #pragma unroll
    for (int j = 0; j < 8; ++j) {                       // C: VGPR j -> M=j (lanes 0-15) / j+8
        int m = j + 8 * half;
        float bias = gsum[m * E_ + 0] * be[0 * D_ + n]
                   + gsum[m * E_ + 1] * be[1 * D_ + n]
                   + gsum[m * E_ + 2] * be[2 * D_ + n];
        u[m * D_ + n] = c[j] + bias;
    }
}

// ---------------- Kernel 7: projector via f32 WMMA: out = (u @ Wp + 1536*bp) / cnt ----------------
// M=16, K=1024, N=4096.
__global__ void k_proj(const float* __restrict__ u, const float* __restrict__ Wp,
                       const float* __restrict__ bp, const float* __restrict__ cntB,
                       float* __restrict__ out) {
    int tid = threadIdx.x;
    int wave = tid >> 5, lane = tid & 31;
    int ml = lane & 15, half = lane >> 4;
    int n0 = (blockIdx.x * 4 + wave) * 16;
    v8f c = {};
    const float* ur = u + ml * D_;
    for (int k = 0; k < D_; k += 4) {
        int ka = k + 2 * half;
        v2f a, bm;
        a.x  = ur[ka];
        a.y  = ur[ka + 1];
        bm.x = Wp[(long)ka * H_ + n0 + ml];
        bm.y = Wp[(long)(ka + 1) * H_ + n0 + ml];
        c = __builtin_amdgcn_wmma_f32_16x16x4_f32(false, a, false, bm,
                                                  (short)0, c, false, false);
    }
    int n = n0 + ml;
    float bpv = (float)TPB_ * bp[n];
    #pragma unroll
    for (int j = 0; j < 8; ++j) {
        int m = j + 8 * half;
        out[m * H_ + n] = (c[j] + bpv) / cntB[m];
    }
}

extern "C" void kernel_launch(void* const* d_in, const int* in_sizes, int n_in,
                              void* d_out, int out_size, void* d_ws, size_t ws_size,
                              hipStream_t stream) {
    (void)in_sizes; (void)n_in; (void)out_size; (void)ws_size;
    const float* x    = (const float*)d_in[0];
    const int*   mask = (const int*)d_in[1];
    const float* wg   = (const float*)d_in[2];
    const float* We   = (const float*)d_in[3];
    const float* be   = (const float*)d_in[4];
    const float* Wp   = (const float*)d_in[5];
    const float* bp   = (const float*)d_in[6];
    float* out = (float*)d_out;

    // workspace layout (elements, all 4B)
    int*   chunkCnt = (int*)d_ws;                          // NCH_*E_ = 288 (zeroed)
    int*   chunkOff = chunkCnt + NCH_ * E_;                // 288
    int*   idxA     = chunkOff + NCH_ * E_;                // 24576
    float* gateA    = (float*)(idxA + S_);                 // 24576
    float* gates3   = gateA + S_;                          // 73728
    float* gateSum  = gates3 + (size_t)S_ * 3;             // 3
    float* cntB     = gateSum + E_;                        // 16
    float* zC       = cntB + B_;                           // 96*3*1024
    float* gsumC    = zC + (size_t)NCH_ * E_ * D_;         // 288
    float* z        = gsumC + NCH_ * E_;                   // 48*1024
    float* gsum     = z + B_ * E_ * D_;                    // 48
    float* u        = gsum + B_ * E_;                      // 16*1024

    hipMemsetAsync(chunkCnt, 0, NCH_ * E_ * sizeof(int), stream);

    k_gate   <<<S_ / 8, 256, 0, stream>>>(x, wg, idxA, gateA, gates3, chunkCnt);
    k_gred   <<<E_,     256, 0, stream>>>(gates3, gateSum);
    k_scan   <<<1,       32, 0, stream>>>(chunkCnt, chunkOff, gateSum, out + B_ * H_);
    k_cnt    <<<B_,     256, 0, stream>>>(mask, cntB);
    k_combine<<<NCH_,   256, 0, stream>>>(x, idxA, gateA, chunkOff, zC, gsumC);
    k_zred   <<<B_ * E_,256, 0, stream>>>(zC, gsumC, z, gsum);
    k_expert <<<B_,     128, 0, stream>>>(z, We, be, gsum, u);      // 64 waves = 64 N-tiles
    k_proj   <<<H_ / 64,128, 0, stream>>>(u, Wp, bp, cntB, out);    // 256 N-tiles
}